// MyModel_87522843561350
// MI455X (gfx1250) — compile-verified
//
#include <hip/hip_runtime.h>
#include <cmath>

#define B_TOTAL 2048
#define T_STEPS 512
#define FEATS   64
#define UNITS   10
#define GATES   30            // 3 * UNITS
#define HPAD    12            // UNITS padded to multiple of 4 for WMMA K

typedef __attribute__((ext_vector_type(2))) float v2f;
typedef __attribute__((ext_vector_type(8))) float v8f;

__global__ __launch_bounds__(32)
void gru_fused_wmma(const float* __restrict__ x,     // [B, T, F]
                    const float* __restrict__ W,     // [F, 3U]
                    const float* __restrict__ Uk,    // [U, 3U]
                    const float* __restrict__ bias,  // [2, 3U]
                    float* __restrict__ out)         // [B, U]
{
    const int lane  = threadIdx.x;     // 0..31 (wave32)
    const int row16 = lane & 15;       // M index within 16-row tile / N column
    const int half  = lane >> 4;       // selects K pair (A/B frags) or M+8 (C/D)
    const int b0    = blockIdx.x * 16; // batch tile base

    // Per-workgroup LDS (single wave per workgroup)
    __shared__ float sS[2][16 * 32];   // [0]=mx dump, [1]=mi dump (row*32 + col)
    __shared__ float sH[2][16 * HPAD]; // ping-pong hidden state, K-padded

    // ---------- Preload B-fragments for input kernel W (16 K-chunks x 2 N-tiles)
    // B 4x16 layout: VGPR0 holds K = kbase + 2*half, VGPR1 holds K+1; N = lane%16.
    v2f wfrag[16][2];
#pragma unroll
    for (int kc = 0; kc < 16; ++kc) {
#pragma unroll
        for (int nt = 0; nt < 2; ++nt) {
            const int col = nt * 16 + row16;
            const int kr  = kc * 4 + 2 * half;
            float f0 = (col < GATES) ? W[(size_t)kr * GATES + col]       : 0.0f;
            float f1 = (col < GATES) ? W[(size_t)(kr + 1) * GATES + col] : 0.0f;
            wfrag[kc][nt] = (v2f){f0, f1};
        }
    }
    // ---------- B-fragments for recurrent kernel U (3 K-chunks, K rows >= 10 are zero)
    v2f ufrag[3][2];
#pragma unroll
    for (int kc = 0; kc < 3; ++kc) {
#pragma unroll
        for (int nt = 0; nt < 2; ++nt) {
            const int col = nt * 16 + row16;
            const int kr  = kc * 4 + 2 * half;
            float f0 = (col < GATES && kr     < UNITS) ? Uk[(size_t)kr * GATES + col]       : 0.0f;
            float f1 = (col < GATES && kr + 1 < UNITS) ? Uk[(size_t)(kr + 1) * GATES + col] : 0.0f;
            ufrag[kc][nt] = (v2f){f0, f1};
        }
    }
    // ---------- Bias broadcast values for accumulator seeding (C layout is row-invariant)
    float bin[2], brec[2];
#pragma unroll
    for (int nt = 0; nt < 2; ++nt) {
        const int col = nt * 16 + row16;
        bin[nt]  = (col < GATES) ? bias[col]         : 0.0f;
        brec[nt] = (col < GATES) ? bias[GATES + col] : 0.0f;
    }

    // ---------- h0 = 0 (both ping-pong buffers, including K padding)
    for (int i = lane; i < 16 * HPAD; i += 32) { sH[0][i] = 0.0f; sH[1][i] = 0.0f; }
    __syncthreads();

    const float* xrow = x + (size_t)(b0 + row16) * T_STEPS * FEATS; // this lane's batch row
    int cur = 0;

    for (int t = 0; t < T_STEPS; ++t) {
        // Seed accumulators with biases: D = x@W + b_in, E = h@U + b_rec
        v8f d0, d1, e0, e1;
#pragma unroll
        for (int i = 0; i < 8; ++i) {
            d0[i] = bin[0];  d1[i] = bin[1];
            e0[i] = brec[0]; e1[i] = brec[1];
        }

        // ---- input GEMM: [16x64] @ [64x32], 16 K-chunks of 4
        const float* xt = xrow + (size_t)t * FEATS;
#pragma unroll
        for (int kc = 0; kc < 16; ++kc) {
            v2f a = *(const v2f*)(xt + kc * 4 + 2 * half); // A frag: K pair for this half
            d0 = __builtin_amdgcn_wmma_f32_16x16x4_f32(false, a, false, wfrag[kc][0],
                                                       (short)0, d0, false, false);
            d1 = __builtin_amdgcn_wmma_f32_16x16x4_f32(false, a, false, wfrag[kc][1],
                                                       (short)0, d1, false, false);
        }
        // ---- recurrent GEMM: [16x12] @ [12x32], 3 K-chunks of 4 (rows 10,11 zero)
#pragma unroll
        for (int kc = 0; kc < 3; ++kc) {
            v2f a = *(const v2f*)(&sH[cur][row16 * HPAD + kc * 4 + 2 * half]);
            e0 = __builtin_amdgcn_wmma_f32_16x16x4_f32(false, a, false, ufrag[kc][0],
                                                       (short)0, e0, false, false);
            e1 = __builtin_amdgcn_wmma_f32_16x16x4_f32(false, a, false, ufrag[kc][1],
                                                       (short)0, e1, false, false);
        }

        // ---- dump accumulators to LDS (C layout: M = i + 8*half, N = row16 [+16])
#pragma unroll
        for (int i = 0; i < 8; ++i) {
            const int m = i + 8 * half;
            sS[0][m * 32 + row16]      = d0[i];
            sS[0][m * 32 + 16 + row16] = d1[i];
            sS[1][m * 32 + row16]      = e0[i];
            sS[1][m * 32 + 16 + row16] = e1[i];
        }
        __syncthreads(); // single-wave WG: barrier ~S_NOP, keeps s_wait_dscnt ordering

        // ---- gate math: lane -> (row = row16, unit = 2*ub + half), 5 passes = 160 states
#pragma unroll
        for (int ub = 0; ub < 5; ++ub) {
            const int u = ub * 2 + half;            // 0..9
            const int r = row16;
            const float mxz = sS[0][r * 32 + u];
            const float miz = sS[1][r * 32 + u];
            const float mxr = sS[0][r * 32 + UNITS + u];
            const float mir = sS[1][r * 32 + UNITS + u];
            const float mxh = sS[0][r * 32 + 2 * UNITS + u];
            const float mih = sS[1][r * 32 + 2 * UNITS + u];
            const float hold = sH[cur][r * HPAD + u];

            const float z  = 1.0f / (1.0f + expf(-(mxz + miz)));
            const float rr = 1.0f / (1.0f + expf(-(mxr + mir)));
            const float hh = tanhf(mxh + rr * mih);
            sH[cur ^ 1][r * HPAD + u] = z * hold + (1.0f - z) * hh;
        }
        __syncthreads();
        cur ^= 1;
    }

    // ---- write h_last: out[b, u]
#pragma unroll
    for (int ub = 0; ub < 5; ++ub) {
        const int u = ub * 2 + half;
        out[(size_t)(b0 + row16) * UNITS + u] = sH[cur][row16 * HPAD + u];
    }
}

extern "C" void kernel_launch(void* const* d_in, const int* in_sizes, int n_in,
                              void* d_out, int out_size, void* d_ws, size_t ws_size,
                              hipStream_t stream) {
    (void)in_sizes; (void)n_in; (void)d_ws; (void)ws_size; (void)out_size;
    const float* x    = (const float*)d_in[0]; // [2048, 512, 64]
    const float* W    = (const float*)d_in[1]; // [64, 30]
    const float* Uk   = (const float*)d_in[2]; // [10, 30]
    const float* bias = (const float*)d_in[3]; // [2, 30]
    float* out = (float*)d_out;                // [2048, 10]

    dim3 grid(B_TOTAL / 16); // 128 batch tiles, one wave32 each
    dim3 block(32);
    gru_fused_wmma<<<grid, block, 0, stream>>>(x, W, Uk, bias, out);
}